// MessagePassing_74612171866795
// MI455X (gfx1250) — compile-verified
//
#include <hip/hip_runtime.h>
#include <hip/hip_bf16.h>

typedef float v2f __attribute__((ext_vector_type(2)));
typedef float v8f __attribute__((ext_vector_type(8)));
typedef unsigned int u32x4 __attribute__((ext_vector_type(4)));
typedef int i32x4 __attribute__((ext_vector_type(4)));
typedef int i32x8 __attribute__((ext_vector_type(8)));

// D = A(16x4 f32) * B(4x16 f32) + C(16x16 f32), native fp32 WMMA on gfx1250.
static __device__ __forceinline__ v8f wmma4(v2f a, v2f b, v8f c) {
  return __builtin_amdgcn_wmma_f32_16x16x4_f32(
      /*neg_a=*/false, a, /*neg_b=*/false, b,
      /*c_mod=*/(short)0, c, /*reuse_a=*/false, /*reuse_b=*/false);
}

static __device__ __forceinline__ v8f vzero8() {
  v8f z;
#pragma unroll
  for (int r = 0; r < 8; ++r) z[r] = 0.0f;
  return z;
}

#define Bb 4
#define Nn 512
#define NF 128
#define EF 64
#define OF 128

// Padded LDS tile row stride produced by the TDM: 64 dwords + 2 pad dwords
// every 32 dwords -> 68 dwords/row; element (m,k) at m*68 + k + 2*(k>>5).
#define TROW 68

// ---------------------------------------------------------------------------
// TDM issue: 2D tile (16 rows x 64 f32) from global 'gaddr' into LDS 'lds_off'
// with padding (pad_interval=32 dwords, pad_amount=2 dwords).
// ---------------------------------------------------------------------------
static __device__ __forceinline__ void tdm_issue_tile(const float* gaddr,
                                                      unsigned int lds_off) {
  unsigned long long ga = (unsigned long long)(uintptr_t)gaddr;
  u32x4 g0;
  g0[0] = 1u;                                   // count=1 valid descriptor
  g0[1] = lds_off;                              // LDS byte address
  g0[2] = (unsigned int)ga;                     // global_addr[31:0]
  g0[3] = ((unsigned int)(ga >> 32) & 0x01FFFFFFu) | (2u << 30);  // addr hi | type=2
  i32x8 g1;
  g1[0] = (2 << 16)        // data_size = 4 bytes
        | (1 << 20)        // pad_enable
        | (4 << 22)        // pad_interval code 4 = 32 dwords
        | (1 << 25);       // pad_amount  code 1 = 2 dwords
  g1[1] = (EF << 16);      // tensor_dim0 = 64 (low 16 in [31:16])
  g1[2] = (Nn << 16);      // tensor_dim0 hi = 0 ; tensor_dim1 low16 = 512
  g1[3] = (EF << 16);      // tensor_dim1 hi = 0 ; tile_dim0 = 64
  g1[4] = 16;              // tile_dim1 = 16 ; tile_dim2 = 0
  g1[5] = EF;              // tensor_dim0_stride low32 = 64
  g1[6] = 0;               // stride hi ; tensor_dim1_stride low
  g1[7] = 0;
  i32x4 z4 = {0, 0, 0, 0};
#if __has_include(<hip/amd_detail/amd_gfx1250_TDM.h>)
  i32x8 z8 = {0, 0, 0, 0, 0, 0, 0, 0};
  __builtin_amdgcn_tensor_load_to_lds(g0, g1, z4, z4, z8, 0);
#else
  __builtin_amdgcn_tensor_load_to_lds(g0, g1, z4, z4, 0);
#endif
}

// ---------------------------------------------------------------------------
// Kernel 1: xp = x @ Wn^T ; mi = xp @ W1^T ; mj = xp @ W2^T   (all 2048x128)
// One wave32 per 16-row M tile. B-fragments are batch-loaded into registers
// (one clause + one wait per 32 WMMAs) so the single wave isn't
// latency-serialized on a recycled operand register.
// ---------------------------------------------------------------------------
__global__ void __launch_bounds__(32)
mp_prep_kernel(const float* __restrict__ x, const float* __restrict__ Wn,
               const float* __restrict__ Wm,
               float* __restrict__ xp_ws, float* __restrict__ mi_ws,
               float* __restrict__ mj_ws) {
  __shared__ float xs[16][132];
  const int lane = threadIdx.x;
  const int n = lane & 15;
  const int g = lane >> 4;
  const int m0 = blockIdx.x * 16;

  v2f afr[32];
#pragma unroll
  for (int kk = 0; kk < 32; ++kk)
    afr[kk] = *(const v2f*)(x + (size_t)(m0 + n) * NF + 4 * kk + 2 * g);

#pragma unroll
  for (int nt = 0; nt < 8; ++nt) {
    const int o0 = nt * 16;
    // Batch-load all 32 B fragments first (single clause, single wait).
    v2f bfr[32];
#pragma unroll
    for (int kk = 0; kk < 32; ++kk)
      bfr[kk] = *(const v2f*)(Wn + (size_t)(o0 + n) * NF + 4 * kk + 2 * g);
    v8f acc = vzero8();
#pragma unroll
    for (int kk = 0; kk < 32; ++kk) acc = wmma4(afr[kk], bfr[kk], acc);
#pragma unroll
    for (int r = 0; r < 8; ++r) {
      const int m = r + 8 * g;
      xs[m][o0 + n] = acc[r];
      xp_ws[(size_t)(m0 + m) * OF + o0 + n] = acc[r];
    }
  }
  __syncthreads();

  // Hoist xp A-fragments once: identical for all 16 mi/mj output tiles.
  v2f afr2[32];
#pragma unroll
  for (int kk = 0; kk < 32; ++kk)
    afr2[kk] = *(const v2f*)&xs[n][4 * kk + 2 * g];

#pragma unroll
  for (int which = 0; which < 2; ++which) {
    float* outp = which ? mj_ws : mi_ws;
    const int coff = which ? OF : 0;
#pragma unroll
    for (int nt = 0; nt < 8; ++nt) {
      const int p0 = nt * 16;
      v2f bfr[32];
#pragma unroll
      for (int kk = 0; kk < 32; ++kk)
        bfr[kk] = *(const v2f*)(Wm + (size_t)(p0 + n) * (3 * OF) + coff +
                                4 * kk + 2 * g);
      v8f acc = vzero8();
#pragma unroll
      for (int kk = 0; kk < 32; ++kk) acc = wmma4(afr2[kk], bfr[kk], acc);
#pragma unroll
      for (int r = 0; r < 8; ++r)
        outp[(size_t)(m0 + r + 8 * g) * OF + p0 + n] = acc[r];
    }
  }
}

// ---------------------------------------------------------------------------
// Kernel 2: fused main pass. One workgroup (4 wave32) per (b,i).
// TDM double-buffers the adj tile into LDS; both GEMMs run on fully
// register-resident operands; adjp streamed out with coalesced b128 stores.
// ---------------------------------------------------------------------------
__global__ void __launch_bounds__(128)
mp_main_kernel(const float* __restrict__ adj, const float* __restrict__ adj_mask,
               const float* __restrict__ We, const float* __restrict__ be,
               const float* __restrict__ Wm, const float* __restrict__ bm,
               const float* __restrict__ xp_ws, const float* __restrict__ mi_ws,
               const float* __restrict__ mj_ws,
               float* __restrict__ out_newx, float* __restrict__ out_adjp,
               float* __restrict__ out_mask) {
  __shared__ float tileBuf[2][16 * TROW];  // TDM-staged adj tiles (padded rows)
  __shared__ float sP[16][132];            // adjp strip
  __shared__ float nbuf[128];
  __shared__ float scnt;

  const int tid = threadIdx.x;
  const int w = tid >> 5;
  const int lane = tid & 31;
  const int n = lane & 15;
  const int g = lane >> 4;
  const int b = blockIdx.x >> 9;
  const int i = blockIdx.x & 511;

  if (tid < 128) nbuf[tid] = 0.0f;
  if (tid == 0) scnt = 0.0f;

  // Loop-invariant weight B-fragments in registers.
  v2f weA[2][16];
  v2f wB[2][32];
#pragma unroll
  for (int t = 0; t < 2; ++t) {
    const int o0 = (w * 2 + t) * 16;
#pragma unroll
    for (int kk = 0; kk < 16; ++kk)
      weA[t][kk] = *(const v2f*)(We + (size_t)(o0 + n) * EF + 4 * kk + 2 * g);
#pragma unroll
    for (int kk = 0; kk < 32; ++kk)
      wB[t][kk] = *(const v2f*)(Wm + (size_t)(o0 + n) * (3 * OF) + 2 * OF +
                                4 * kk + 2 * g);
  }

  const size_t rowbase = ((size_t)b * Nn + i) * Nn;
  const unsigned int ldsT0 = (unsigned int)(uintptr_t)&tileBuf[0][0];
  const unsigned int ldsT1 = (unsigned int)(uintptr_t)&tileBuf[1][0];

  // Prefetch first tile.
  if (w == 0) tdm_issue_tile(adj + rowbase * EF, ldsT0);

  float nb0 = 0.0f, nb1 = 0.0f;
  float cl = 0.0f;

  for (int j0 = 0; j0 < Nn; j0 += 16) {
    const int cur = (j0 >> 4) & 1;
    // Issue next tile; guarantee current tile landed.
    if (w == 0) {
      if (j0 + 16 < Nn) {
        tdm_issue_tile(adj + (rowbase + j0 + 16) * EF, cur ? ldsT0 : ldsT1);
        __builtin_amdgcn_s_wait_tensorcnt(1);
      } else {
        __builtin_amdgcn_s_wait_tensorcnt(0);
      }
    }
    __syncthreads();  // B1: tile[cur] ready; prev iter's sP readers all done

    // ---- A fragments for adj tile from padded LDS stage ----
    const float* tb = cur ? &tileBuf[1][0] : &tileBuf[0][0];
    v2f afr[16];
#pragma unroll
    for (int kk = 0; kk < 16; ++kk) {
      const int k = 4 * kk + 2 * g;
      afr[kk] = *(const v2f*)(tb + n * TROW + k + 2 * (k >> 5));
    }

    // ---- adjp = adj @ We^T + be : pure register WMMA, stage into sP ----
#pragma unroll
    for (int t = 0; t < 2; ++t) {
      const int o0 = (w * 2 + t) * 16;
      const float bev = be[o0 + n];
      v8f acc;
#pragma unroll
      for (int r = 0; r < 8; ++r) acc[r] = bev;
#pragma unroll
      for (int kk = 0; kk < 16; ++kk) acc = wmma4(afr[kk], weA[t][kk], acc);
#pragma unroll
      for (int r = 0; r < 8; ++r) sP[r + 8 * g][o0 + n] = acc[r];
    }

    // mask copy + count (wave 0)
    if (w == 0 && lane < 16) {
      const float mv = adj_mask[rowbase + j0 + lane];
      out_mask[rowbase + j0 + lane] = mv;
      cl += mv;
    }
    __syncthreads();  // B2: sP complete; tile[cur] reads complete

    // ---- coalesced adjp store: 4 x b128 per thread (512B / instruction) ----
#pragma unroll
    for (int c = 0; c < 4; ++c) {
      const int lin = c * 512 + tid * 4;
      const int m = lin >> 7;
      const int o = lin & 127;
      const float4 vv = *(const float4*)&sP[m][o];
      *(float4*)&out_adjp[(rowbase + j0 + m) * OF + o] = vv;
    }

    // per-lane masks (m = r + 8*g)
    float mk[8];
#pragma unroll
    for (int r = 0; r < 8; ++r) mk[r] = adj_mask[rowbase + j0 + r + 8 * g];

    // ---- me = adjp @ W3^T : hoist A-frags once, reuse for both tiles ----
    v2f afr2[32];
#pragma unroll
    for (int kk = 0; kk < 32; ++kk)
      afr2[kk] = *(const v2f*)&sP[n][4 * kk + 2 * g];

#pragma unroll
    for (int t = 0; t < 2; ++t) {
      const int o0 = (w * 2 + t) * 16;
      v8f macc = vzero8();
#pragma unroll
      for (int kk = 0; kk < 32; ++kk) macc = wmma4(afr2[kk], wB[t][kk], macc);
      float part = 0.0f;
#pragma unroll
      for (int r = 0; r < 8; ++r) {
        const int m = r + 8 * g;
        const float mjv = mj_ws[((size_t)b * Nn + j0 + m) * OF + o0 + n];
        part += mk[r] * (macc[r] + mjv);
      }
      if (t == 0) nb0 += part; else nb1 += part;
    }
    // no trailing barrier needed: next iter's B1 orders sP reuse,
    // and tile[next] is the other buffer.
  }

  atomicAdd(&nbuf[(w * 2 + 0) * 16 + n], nb0);
  atomicAdd(&nbuf[(w * 2 + 1) * 16 + n], nb1);
  if (w == 0) atomicAdd(&scnt, cl);
  __syncthreads();

  if (tid < 128) {
    const size_t xi = ((size_t)b * Nn + i) * OF + tid;
    out_newx[xi] = xp_ws[xi] + scnt * (mi_ws[xi] + bm[tid]) + nbuf[tid];
  }
}

// ---------------------------------------------------------------------------
extern "C" void kernel_launch(void* const* d_in, const int* in_sizes, int n_in,
                              void* d_out, int out_size, void* d_ws,
                              size_t ws_size, hipStream_t stream) {
  const float* x        = (const float*)d_in[0];
  const float* adj      = (const float*)d_in[1];
  const float* adj_mask = (const float*)d_in[2];
  const float* Wn       = (const float*)d_in[3];
  const float* We       = (const float*)d_in[4];
  const float* be       = (const float*)d_in[5];
  const float* Wm       = (const float*)d_in[6];
  const float* bm       = (const float*)d_in[7];

  float* xp_ws = (float*)d_ws;
  float* mi_ws = xp_ws + (size_t)Bb * Nn * OF;
  float* mj_ws = mi_ws + (size_t)Bb * Nn * OF;

  float* out_newx = (float*)d_out;
  float* out_adjp = out_newx + (size_t)Bb * Nn * OF;
  float* out_mask = out_adjp + (size_t)Bb * Nn * Nn * OF;

  mp_prep_kernel<<<(Bb * Nn) / 16, 32, 0, stream>>>(x, Wn, Wm, xp_ws, mi_ws,
                                                    mj_ws);
  mp_main_kernel<<<Bb * Nn, 128, 0, stream>>>(adj, adj_mask, We, be, Wm, bm,
                                              xp_ws, mi_ws, mj_ws, out_newx,
                                              out_adjp, out_mask);
}